// GM_GCN_81028853006975
// MI455X (gfx1250) — compile-verified
//
#include <hip/hip_runtime.h>

typedef __attribute__((ext_vector_type(2))) float v2f;
typedef __attribute__((ext_vector_type(8))) float v8f;

#define NNODES 100000
#define NEDGES 1250000

// ---------------- utility kernels ----------------

__global__ void fill_kernel(float* __restrict__ p, long long n, float v) {
    long long i = (long long)blockIdx.x * blockDim.x + threadIdx.x;
    if (i < n) p[i] = v;
}

__global__ void deg_kernel(const int* __restrict__ col, float* __restrict__ deg, int E) {
    int e = blockIdx.x * blockDim.x + threadIdx.x;
    if (e < E) atomicAdd(&deg[col[e]], 1.0f);
}

__global__ void rsqrt_kernel(float* __restrict__ d, int n) {
    int i = blockIdx.x * blockDim.x + threadIdx.x;
    if (i < n) {
        float x = d[i];
        d[i] = (x > 0.0f) ? rsqrtf(x) : 0.0f;
    }
}

// ---------------- WMMA fp32 GEMM: Y[M,Nout] = X[M,K] @ W[K,Nout] (+bias) ----------------
// W (<=32KB) is staged transposed+zero-padded into LDS once per block; each wave then
// computes a full 16-row x NOUTPAD strip: one A fragment (global_load_b64) feeds
// NOUTPAD/16 v_wmma_f32_16x16x4_f32 ops per K-step, with B fragments from LDS
// (ds_load_b64, bank-conflict-free via STRIDE = K+4).
//
// Fragment layouts (ISA 7.12.2, 32-bit data, wave32):
//   A 16x4: lanes 0-15 row M=l hold K={k,k+1}; lanes 16-31 hold K={k+2,k+3}
//   B 4x16: lanes 0-15 col N=l hold rows K={k,k+1}; lanes 16-31 rows K={k+2,k+3}
//   C/D 16x16: VGPR v -> row v + 8*half, col = l
template <int K, int NOUTPAD>
__global__ __launch_bounds__(256) void gemm_wmma_kernel(
    const float* __restrict__ X, const float* __restrict__ W,
    const float* __restrict__ bias, float* __restrict__ Y,
    int M, int Nout, int addBias)
{
    constexpr int STRIDE = K + 4;
    constexpr int NT = NOUTPAD / 16;
    __shared__ float Wt[NOUTPAD * STRIDE];

    // Stage W transposed into LDS, zero-padding columns >= Nout.
    for (int idx = threadIdx.x; idx < NOUTPAD * K; idx += 256) {
        int n = idx / K;
        int k = idx - n * K;
        Wt[n * STRIDE + k] = (n < Nout) ? W[(size_t)k * Nout + n] : 0.0f;
    }
    __syncthreads();

    const int lane  = threadIdx.x & 31;
    const int strip = blockIdx.x * 8 + (threadIdx.x >> 5);  // 16-row strip index
    const int Mtiles = M >> 4;                               // M is a multiple of 16
    if (strip >= Mtiles) return;                             // wave-uniform

    const int half = lane >> 4;     // 0: lanes 0-15, 1: lanes 16-31
    const int l    = lane & 15;
    const int row0 = strip << 4;

    const float* __restrict__ ap  = X + (size_t)(row0 + l) * K + 2 * half;
    const float* __restrict__ bp0 = &Wt[l * STRIDE + 2 * half];

    v8f acc[NT];
#pragma unroll
    for (int t = 0; t < NT; ++t) acc[t] = {};

#pragma unroll 4
    for (int k = 0; k < K; k += 4) {
        v2f a;
        a.x = ap[k];
        a.y = ap[k + 1];
#pragma unroll
        for (int t = 0; t < NT; ++t) {
            v2f b;
            const float* bp = bp0 + t * 16 * STRIDE;
            b.x = bp[k];
            b.y = bp[k + 1];
            acc[t] = __builtin_amdgcn_wmma_f32_16x16x4_f32(
                false, a, false, b, (short)0, acc[t], false, false);
        }
    }

#pragma unroll
    for (int t = 0; t < NT; ++t) {
        int c = t * 16 + l;
        if (c < Nout) {
            float bv = addBias ? bias[c] : 0.0f;
#pragma unroll
            for (int v = 0; v < 8; ++v) {
                int r = row0 + v + 8 * half;
                Y[(size_t)r * Nout + c] = acc[t][v] + bv;
            }
        }
    }
}

// ---------------- edge scatter: AGG[col] += H[row] * (dis[row]*dis[col]) ----------------
// F = 64 features; 16 lanes per edge, float4 per lane. Atomic adds resolve in L2
// (25.6MB aggregation buffer << 192MB L2).
__global__ void scatter_kernel(const float* __restrict__ H, const int* __restrict__ row,
                               const int* __restrict__ col, const float* __restrict__ dis,
                               float* __restrict__ AGG, int E) {
    long long gid = (long long)blockIdx.x * blockDim.x + threadIdx.x;
    int e  = (int)(gid >> 4);
    int f4 = (int)(gid & 15) << 2;
    if (e >= E) return;
    int r = row[e];
    int c = col[e];
    float nrm = dis[r] * dis[c];
    const float4 h = *(const float4*)(H + (size_t)r * 64 + f4);
    float* out = AGG + (size_t)c * 64 + f4;
    atomicAdd(out + 0, h.x * nrm);
    atomicAdd(out + 1, h.y * nrm);
    atomicAdd(out + 2, h.z * nrm);
    atomicAdd(out + 3, h.w * nrm);
}

// ---------------- combine: agg = ReLU(agg + h*dis^2 + b)  (adds the self-loop term) ----
__global__ void combine_kernel(float* __restrict__ agg, const float* __restrict__ htmp,
                               const float* __restrict__ dis, const float* __restrict__ bias,
                               int n /* NNODES*64 */) {
    int i = blockIdx.x * blockDim.x + threadIdx.x;
    if (i >= n) return;
    int node = i >> 6;
    int f    = i & 63;
    float d = dis[node];
    float v = agg[i] + htmp[i] * d * d + bias[f];
    agg[i] = (v > 0.0f) ? v : 0.0f;
}

// ---------------- launcher ----------------

extern "C" void kernel_launch(void* const* d_in, const int* in_sizes, int n_in,
                              void* d_out, int out_size, void* d_ws, size_t ws_size,
                              hipStream_t stream) {
    const float* x    = (const float*)d_in[0];              // [N,128]
    const int*   ei   = (const int*)d_in[1];                // [2,E] flattened
    const float* W1   = (const float*)d_in[2];              // [128,64]
    const float* b1   = (const float*)d_in[3];              // [64]
    const float* W2   = (const float*)d_in[4];              // [64,64]
    const float* b2   = (const float*)d_in[5];              // [64]
    const float* Wout = (const float*)d_in[6];              // [64,40]
    const float* bout = (const float*)d_in[7];              // [40]
    float* out = (float*)d_out;                             // [N,40]

    const int* row = ei;                                    // first E entries
    const int* col = ei + NEDGES;                           // next E entries

    float* dis = (float*)d_ws;                              // [N]
    float* A   = dis + NNODES;                              // [N,64] gemm temp
    float* B   = A + (size_t)NNODES * 64;                   // [N,64] agg / H1
    float* C   = B + (size_t)NNODES * 64;                   // [N,64] agg / H2
    (void)ws_size; (void)n_in; (void)in_sizes; (void)out_size;

    const int T = 256;
    const long long NF = (long long)NNODES * 64;
    const int Mtiles = NNODES / 16;                         // 6250
    const int gemmBlocks = (Mtiles + 7) / 8;                // 8 waves (strips) per block

    // normalization: deg (with self loop) -> dis = rsqrt(deg)
    fill_kernel<<<(NNODES + T - 1) / T, T, 0, stream>>>(dis, NNODES, 1.0f);
    deg_kernel<<<(NEDGES + T - 1) / T, T, 0, stream>>>(col, dis, NEDGES);
    rsqrt_kernel<<<(NNODES + T - 1) / T, T, 0, stream>>>(dis, NNODES);

    // ---- layer 1: A = X @ W1 ; B = ReLU(scatter(A) + A*dis^2 + b1)
    gemm_wmma_kernel<128, 64><<<gemmBlocks, T, 0, stream>>>(x, W1, nullptr, A, NNODES, 64, 0);
    fill_kernel<<<(int)((NF + T - 1) / T), T, 0, stream>>>(B, NF, 0.0f);
    scatter_kernel<<<(int)(((long long)NEDGES * 16 + T - 1) / T), T, 0, stream>>>(A, row, col, dis, B, NEDGES);
    combine_kernel<<<(int)((NF + T - 1) / T), T, 0, stream>>>(B, A, dis, b1, (int)NF);

    // ---- layer 2: A = B @ W2 ; C = ReLU(scatter(A) + A*dis^2 + b2)
    gemm_wmma_kernel<64, 64><<<gemmBlocks, T, 0, stream>>>(B, W2, nullptr, A, NNODES, 64, 0);
    fill_kernel<<<(int)((NF + T - 1) / T), T, 0, stream>>>(C, NF, 0.0f);
    scatter_kernel<<<(int)(((long long)NEDGES * 16 + T - 1) / T), T, 0, stream>>>(A, row, col, dis, C, NEDGES);
    combine_kernel<<<(int)((NF + T - 1) / T), T, 0, stream>>>(C, A, dis, b2, (int)NF);

    // ---- output layer: out = C @ Wout + bout   (Nout=40 zero-padded to 48 in LDS)
    gemm_wmma_kernel<64, 48><<<gemmBlocks, T, 0, stream>>>(C, Wout, bout, out, NNODES, 40, 1);
}